// ExodusNeuron_7516192769003
// MI455X (gfx1250) — compile-verified
//
#include <hip/hip_runtime.h>
#include <cstdint>

// Reference shapes: x (B=32, T=2048, N=512, 1) f32; weight (1,1) f32.
// out: spikes, same shape as x, f32.
constexpr int B_ = 32;
constexpr int T_ = 2048;
constexpr int N_ = 512;

constexpr float ALPHA = 0.95122942450071400910f; // exp(-1/20)
constexpr float VTH   = 1.0f;

constexpr int NTILE       = 64;                    // neurons per block == threads per block
constexpr int NTILES      = N_ / NTILE;            // 8 tiles cover N=512
constexpr int CT          = 128;                   // timesteps staged per chunk
constexpr int NCHUNK      = T_ / CT;               // 16
constexpr int ROW_BYTES   = NTILE * 4;             // 256 B of x per timestep per block
constexpr int SEGS_ROW    = ROW_BYTES / 16;        // 16 b128 segments per row
constexpr int SEGS_CHUNK  = CT * SEGS_ROW;         // 2048 segments per chunk
constexpr int SEGS_THREAD = SEGS_CHUNK / NTILE;    // 32 async issues per thread per chunk
constexpr int BUF_BYTES   = CT * NTILE * 4;        // 32 KiB per LDS buffer

// One wave-level async bulk copy: 32 lanes x 16B = 512B global -> LDS,
// tracked on ASYNCcnt (CDNA5 async path). GVS addressing: saddr(64b) + vaddr(32b).
__device__ __forceinline__ void async_b128_to_lds(uint32_t lds_off,
                                                  const float* sbase,
                                                  uint32_t byte_off) {
  asm volatile("global_load_async_to_lds_b128 %0, %1, %2"
               :
               : "v"(lds_off), "v"(byte_off), "s"(sbase)
               : "memory");
}

// Stage one [CT x NTILE] f32 tile of x into an LDS buffer.
// Segment s covers bytes [s*16, s*16+16) of the tile (row-major [t][n]).
__device__ __forceinline__ void issue_chunk(const float* gbase, int t0,
                                            uint32_t lds_buf, int tid) {
#pragma unroll
  for (int j = 0; j < SEGS_THREAD; ++j) {
    const int s   = j * NTILE + tid;            // 0 .. SEGS_CHUNK-1
    const int row = s >> 4;                     // s / SEGS_ROW (timestep within chunk)
    const int col = s & (SEGS_ROW - 1);         // 16B segment within the 256B row
    const uint32_t voff = (uint32_t)((t0 + row) * (N_ * 4) + col * 16);
    async_b128_to_lds(lds_buf + (uint32_t)s * 16u, gbase, voff);
  }
}

__global__ __launch_bounds__(NTILE, 1)
void exodus_lif_scan(const float* __restrict__ x,
                     const float* __restrict__ weight,
                     float* __restrict__ out) {
  extern __shared__ float smem[];               // 2 x BUF_BYTES, double buffered

  const int tid   = threadIdx.x;                // neuron within tile (one chain/thread)
  const int b     = blockIdx.x >> 3;            // batch
  const int ntile = blockIdx.x & (NTILES - 1);  // 8 tiles of 64 neurons

  const float w = weight[0];                    // uniform -> scalar load

  const float* gbase = x + ((size_t)b * T_) * N_ + (size_t)ntile * NTILE;
  float*       obase = out + ((size_t)b * T_) * N_ + (size_t)ntile * NTILE + tid;

  // Low 32 bits of the flat address of a __shared__ object == LDS byte offset,
  // which is what the async-load VDST operand wants.
  const uint32_t lds0 = (uint32_t)(uintptr_t)(void*)smem;

  float vsyn = 0.0f;
  float vmem = 0.0f;

  // Prologue: fill buffer 0 with chunk 0.
  issue_chunk(gbase, 0, lds0, tid);

  for (int c = 0; c < NCHUNK; ++c) {
    if (c + 1 < NCHUNK) {
      // Prefetch next chunk into the other buffer, then wait until only the
      // 32 just-issued loads remain outstanding => chunk c fully landed
      // (ASYNCcnt loads complete in order within a wave).
      issue_chunk(gbase, (c + 1) * CT, lds0 + (uint32_t)((c + 1) & 1) * BUF_BYTES, tid);
      asm volatile("s_wait_asynccnt 32" ::: "memory");
    } else {
      asm volatile("s_wait_asynccnt 0" ::: "memory");
    }
    __syncthreads();   // every wave's async data for chunk c is in LDS

    const float* buf = smem + (size_t)(c & 1) * (BUF_BYTES / 4);

#pragma unroll 8
    for (int t = 0; t < CT; ++t) {
      const float xv = buf[t * NTILE + tid];    // bank-conflict-free LDS read
      // ExpLeak:  v_syn = alpha*v_syn + w*x   (w*xv is off the critical chain)
      vsyn = __builtin_fmaf(ALPHA, vsyn, w * xv);
      // LIF:      v_mem = alpha*v_mem + v_syn
      vmem = __builtin_fmaf(ALPHA, vmem, vsyn);
      // SingleSpike + MembraneSubtract: cmp and sub both hang off the fma,
      // cndmask merges -> 3-deep dependent VALU chain per step.
      const float spk = (vmem >= VTH) ? 1.0f : 0.0f;
      vmem -= spk;
      // Streaming output, zero reuse: non-temporal coalesced store.
      __builtin_nontemporal_store(spk, obase + (size_t)(c * CT + t) * N_);
    }
    __syncthreads();   // buffer (c&1) free for the chunk c+2 prefetch
  }
}

extern "C" void kernel_launch(void* const* d_in, const int* in_sizes, int n_in,
                              void* d_out, int out_size, void* d_ws, size_t ws_size,
                              hipStream_t stream) {
  (void)in_sizes; (void)n_in; (void)d_ws; (void)ws_size; (void)out_size;
  const float* x   = (const float*)d_in[0];
  const float* wgt = (const float*)d_in[1];
  float*       out = (float*)d_out;

  dim3 grid(B_ * NTILES);         // 32 * 8 = 256 blocks (512 wave32 waves total)
  dim3 block(NTILE);              // 64 threads = 2 waves
  const size_t shmem = 2u * (size_t)BUF_BYTES;  // 64 KiB double buffer
  hipLaunchKernelGGL(exodus_lif_scan, grid, block, shmem, stream, x, wgt, out);
}